// LearnableDataArugmentation_64802466562527
// MI455X (gfx1250) — compile-verified
//
#include <hip/hip_runtime.h>

typedef __attribute__((ext_vector_type(2))) float v2f;
typedef __attribute__((ext_vector_type(8))) float v8f;

namespace {
constexpr int   kN     = 2048;
constexpr int   kD     = 64;
constexpr int   kIter  = 409;          // int(0.2 * 2048)
constexpr float kTemp  = 1e-4f;
constexpr float kNegInf = -3.402823466e38f;
}

// -----------------------------------------------------------------------------
// Kernel 1: H = x @ W^T + b  via V_WMMA_F32_16X16X4_F32 (exact fp32 matmul).
// Stored transposed (column-major): Hc[c * kN + n], so the sharpen kernel reads
// each column contiguously.
// One wave per 16x16 output tile: blockIdx.x = row tile (128), wave-in-block =
// column tile (4). 16 K-steps of 4.
// -----------------------------------------------------------------------------
__global__ __launch_bounds__(128) void mask_proj_wmma(
    const float* __restrict__ x, const float* __restrict__ W,
    const float* __restrict__ bias, float* __restrict__ Hc) {
  const int lane = threadIdx.x & 31;
  const int wv   = threadIdx.x >> 5;   // 0..3 : 16-column tile
  const int hi   = lane >> 4;          // half-wave select (0/1)
  const int l16  = lane & 15;
  const int m0   = blockIdx.x * 16;    // row-tile base (n dimension)
  const int c0   = wv * 16;            // col-tile base (feature dimension)

  // A (x tile, 16x4 f32): lane L holds row m0 + (L&15); VGPR v holds
  // K = kbase + v + 2*(L>>4).
  const float* xr = x + (m0 + l16) * kD;
  // B (W^T tile, 4x16 f32): B[k][n] = W[c0+n][k]; lane L holds column
  // n = L&15; VGPR v holds K = kbase + v + 2*(L>>4).
  const float* wr = W + (c0 + l16) * kD;

  v8f acc = {};
#pragma unroll
  for (int kk = 0; kk < 16; ++kk) {
    const int k0 = kk * 4 + 2 * hi;
    v2f a, b;
    a.x = xr[k0];
    a.y = xr[k0 + 1];
    b.x = wr[k0];
    b.y = wr[k0 + 1];
    acc = __builtin_amdgcn_wmma_f32_16x16x4_f32(
        false, a, false, b, (short)0, acc, false, false);
  }

  // C/D layout: VGPR i, lanes 0-15 -> M=i, lanes 16-31 -> M=i+8; N = lane&15.
  const int   col = c0 + l16;
  const float bv  = bias[col];
#pragma unroll
  for (int i = 0; i < 8; ++i) {
    const int m = m0 + i + 8 * hi;
    Hc[col * kN + m] = acc[i] + bv;
  }
}

// -----------------------------------------------------------------------------
// Kernel 2: sharpen. One block (256 threads = 8 waves) per feature column.
// Each thread owns 8 rows in registers; 409 sequential rounds of
// {masked max-reduce, exp, sum-reduce, h_hat update}. Wave reduction via
// __shfl_xor (wave32), cross-wave via LDS; 2 barriers / iteration.
//
// Bit-faithfulness notes vs. the fp32 reference:
//  * unmasked: m_hat = log(1.0f + 1e-15f) = log(1.0f) = 0 exactly, so
//    z = (-h)/1e-4f (IEEE div), precomputed once (h never changes).
//  * masked:   z is shifted by log(1e-15)/1e-4 ~ -3.45e5 -> never the column
//    max, exp(z - max) underflows to exactly 0, and y*m = 0 regardless.
//    So masked lanes contribute nothing: implemented as selects.
// -----------------------------------------------------------------------------
__global__ __launch_bounds__(256) void sharpen_kernel(
    const float* __restrict__ Hc, float* __restrict__ out) {
  const int c    = blockIdx.x;        // column 0..63
  const int tid  = threadIdx.x;       // 0..255
  const int lane = tid & 31;
  const int wv   = tid >> 5;          // 0..7

  __shared__ float smax[8];
  __shared__ float ssum[8];

  float zu[8];      // (-h)/temp for unmasked entries (constant)
  float hhat[8];
  bool  masked[8];

#pragma unroll
  for (int j = 0; j < 8; ++j) {
    const float h = Hc[c * kN + j * 256 + tid];   // coalesced column read
    zu[j]     = (-h) / kTemp;
    hhat[j]   = 0.0f;
    masked[j] = false;
  }

  for (int it = 0; it < kIter; ++it) {
    // ---- max over column (masked entries can never win) ----
    float lmax = kNegInf;
#pragma unroll
    for (int j = 0; j < 8; ++j)
      lmax = fmaxf(lmax, masked[j] ? kNegInf : zu[j]);
#pragma unroll
    for (int off = 16; off >= 1; off >>= 1)
      lmax = fmaxf(lmax, __shfl_xor(lmax, off, 32));
    if (lane == 0) smax[wv] = lmax;
    __syncthreads();
    float gmax = smax[0];
#pragma unroll
    for (int w = 1; w < 8; ++w) gmax = fmaxf(gmax, smax[w]);

    // ---- exp + sum (masked -> exactly 0, selected, never inf*0) ----
    float e[8];
    float lsum = 0.0f;
#pragma unroll
    for (int j = 0; j < 8; ++j) {
      const float ev = expf(zu[j] - gmax);
      e[j] = masked[j] ? 0.0f : ev;
      lsum += e[j];
    }
#pragma unroll
    for (int off = 16; off >= 1; off >>= 1)
      lsum += __shfl_xor(lsum, off, 32);
    if (lane == 0) ssum[wv] = lsum;
    __syncthreads();
    float gsum = 0.0f;
#pragma unroll
    for (int w = 0; w < 8; ++w) gsum += ssum[w];

    // ---- h_hat += y * m ; re-derive mask (h_hat == 1.0 exactly) ----
#pragma unroll
    for (int j = 0; j < 8; ++j) {
      if (!masked[j]) {
        hhat[j] += e[j] / gsum;
        masked[j] = (hhat[j] == 1.0f);
      }
    }
  }

  // out is [N, D] row-major: out[n*kD + c] = 1 - h_hat
#pragma unroll
  for (int j = 0; j < 8; ++j)
    out[(j * 256 + tid) * kD + c] = 1.0f - hhat[j];
}

// -----------------------------------------------------------------------------
extern "C" void kernel_launch(void* const* d_in, const int* in_sizes, int n_in,
                              void* d_out, int out_size, void* d_ws, size_t ws_size,
                              hipStream_t stream) {
  const float* x    = (const float*)d_in[0];   // [2048, 64]
  const float* W    = (const float*)d_in[1];   // [64, 64]
  const float* bias = (const float*)d_in[2];   // [64]
  float*       out  = (float*)d_out;           // [2048, 64]
  float*       Hc   = (float*)d_ws;            // [64][2048] column-major H

  (void)in_sizes; (void)n_in; (void)out_size; (void)ws_size;

  mask_proj_wmma<<<kN / 16, 128, 0, stream>>>(x, W, bias, Hc);
  sharpen_kernel<<<kD, 256, 0, stream>>>(Hc, out);
}